// ClusteredAttention_52441550684839
// MI455X (gfx1250) — compile-verified
//
#include <hip/hip_runtime.h>
#include <math.h>

typedef __attribute__((ext_vector_type(2))) float v2f;
typedef __attribute__((ext_vector_type(4))) float v4f;
typedef __attribute__((ext_vector_type(8))) float v8f;

#define B_  4
#define L_  512
#define V_  32
#define D_  64
#define MT  32          // rows per workgroup (two 16-row WMMA tiles)
#define SCALE 0.125f    // 1/sqrt(64)
#define LDSS (L_ + 4)   // padded score-row stride (516): bank stride 4/row

// ---------------------------------------------------------------------------
// Kernel 1: S[b,j,d] = sum_v key[b,j,v,d]
// ---------------------------------------------------------------------------
__global__ __launch_bounds__(256)
void sum_key_kernel(const float* __restrict__ key, float* __restrict__ sumv) {
    int idx = blockIdx.x * blockDim.x + threadIdx.x;   // over B*L*D
    if (idx >= B_ * L_ * D_) return;
    int d  = idx % D_;
    int bl = idx / D_;
    const float* p = key + (size_t)bl * (V_ * D_) + d;
    float s = 0.f;
#pragma unroll
    for (int v = 0; v < V_; ++v) s += p[v * D_];
    sumv[idx] = s;
}

// ---------------------------------------------------------------------------
// Kernel 2: one wave per (b, head k, 32-row i-tile).
//   phase 1: scores(32x512) = Q(32x64) * S^T(64x512)   [2 WMMAs per B-frag]
//   phase 2: masked softmax, one lane per row, float4 LDS passes
//   phase 3: O(32x64) = P(32x512) * V(512x64)          [8 WMMAs per 8 loads]
// ---------------------------------------------------------------------------
__global__ __launch_bounds__(32)
void attn_kernel(const float* __restrict__ q,
                 const float* __restrict__ sumv,
                 const float* __restrict__ val,
                 const int*   __restrict__ lab,
                 float*       __restrict__ out) {
    __shared__ float smem[MT * LDSS];   // 32 x 516 fp32 = 66 KB
    __shared__ float labf[L_];          // labels as float for vector compare
    __shared__ float rowsum[MT];

    const int blk   = blockIdx.x;
    const int itile = blk % (L_ / MT);  // 16 tiles
    const int bk    = blk / (L_ / MT);
    const int k     = bk % V_;          // head index
    const int b     = bk / V_;

    const int lane = threadIdx.x;       // 0..31, wave32, all active
    const int hlf  = lane >> 4;         // 0/1
    const int l16  = lane & 15;
    const int i0   = itile * MT;

    for (int t = lane; t < L_; t += 32) labf[t] = (float)lab[b * L_ + t];
    __syncthreads();

    // ---- preload Q A-fragments for both 16-row tiles ----------------------
    v2f qa0[16], qa1[16];
    {
        const float* q0 =
            q + (((size_t)(b * L_ + i0 + l16) * V_) + k) * D_ + hlf * 2;
        const float* q1 = q0 + (size_t)16 * V_ * D_;
#pragma unroll
        for (int s = 0; s < 16; ++s) {
            qa0[s] = *(const v2f*)(q0 + s * 4);
            qa1[s] = *(const v2f*)(q1 + s * 4);
        }
    }

    // ---- phase 1: scores -> LDS ------------------------------------------
    const float* sb = sumv + (size_t)b * L_ * D_;
    for (int jt = 0; jt < L_ / 16; ++jt) {
        v8f c0 = {}, c1 = {};
        const float* srow = sb + (size_t)(jt * 16 + l16) * D_ + hlf * 2;
#pragma unroll
        for (int s = 0; s < 16; ++s) {
            v2f bf = *(const v2f*)(srow + s * 4);
            c0 = __builtin_amdgcn_wmma_f32_16x16x4_f32(
                     false, qa0[s], false, bf, (short)0, c0, false, false);
            c1 = __builtin_amdgcn_wmma_f32_16x16x4_f32(
                     false, qa1[s], false, bf, (short)0, c1, false, false);
        }
#pragma unroll
        for (int r = 0; r < 8; ++r) {
            smem[(r + hlf * 8) * LDSS + jt * 16 + l16]      = c0[r];
            smem[(16 + r + hlf * 8) * LDSS + jt * 16 + l16] = c1[r];
        }
    }
    __syncthreads();

    // ---- phase 2: masked softmax; lane owns row = lane --------------------
    {
        const int row = lane;
        const float myLab = labf[i0 + row];
        float* srow = &smem[row * LDSS];

        // pass A: apply cluster mask in place, track row max (vectorized)
        v4f m4 = {-INFINITY, -INFINITY, -INFINITY, -INFINITY};
        for (int c = 0; c < L_; c += 4) {
            v4f s  = *(v4f*)(srow + c);
            v4f lv = *(const v4f*)(&labf[c]);
            s.x = (lv.x == myLab) ? s.x : -INFINITY;
            s.y = (lv.y == myLab) ? s.y : -INFINITY;
            s.z = (lv.z == myLab) ? s.z : -INFINITY;
            s.w = (lv.w == myLab) ? s.w : -INFINITY;
            *(v4f*)(srow + c) = s;
            m4.x = fmaxf(m4.x, s.x);  m4.y = fmaxf(m4.y, s.y);
            m4.z = fmaxf(m4.z, s.z);  m4.w = fmaxf(m4.w, s.w);
        }
        const float m = fmaxf(fmaxf(m4.x, m4.y), fmaxf(m4.z, m4.w));

        // pass B: exponentiate (masked entries are -inf -> 0), accumulate sum
        v4f s4 = {0.f, 0.f, 0.f, 0.f};
        for (int c = 0; c < L_; c += 4) {
            v4f s = *(v4f*)(srow + c);
            v4f p;
            p.x = __expf(SCALE * (s.x - m));
            p.y = __expf(SCALE * (s.y - m));
            p.z = __expf(SCALE * (s.z - m));
            p.w = __expf(SCALE * (s.w - m));
            *(v4f*)(srow + c) = p;
            s4 += p;
        }
        rowsum[row] = (s4.x + s4.y) + (s4.z + s4.w);
    }
    __syncthreads();

    // ---- phase 3: O = P * V (B-frags shared by both row tiles) ------------
    v8f acc[2][4] = {{{}, {}, {}, {}}, {{}, {}, {}, {}}};
    const float* vb = val + ((size_t)(b * L_) * V_ + k) * D_;  // + j*V_*D_ + d
    for (int k0 = 0; k0 < L_; k0 += 4) {
        const int j0 = k0 + hlf * 2;
        const v2f af0 = *(const v2f*)(&smem[l16 * LDSS + j0]);
        const v2f af1 = *(const v2f*)(&smem[(16 + l16) * LDSS + j0]);
        const float* v0 = vb + (size_t)j0 * (V_ * D_);
        const float* v1 = v0 + (V_ * D_);
#pragma unroll
        for (int n = 0; n < 4; ++n) {
            v2f bf;
            bf.x = v0[n * 16 + l16];
            bf.y = v1[n * 16 + l16];
            acc[0][n] = __builtin_amdgcn_wmma_f32_16x16x4_f32(
                            false, af0, false, bf, (short)0, acc[0][n], false, false);
            acc[1][n] = __builtin_amdgcn_wmma_f32_16x16x4_f32(
                            false, af1, false, bf, (short)0, acc[1][n], false, false);
        }
    }

    // ---- epilogue: normalize and store -----------------------------------
#pragma unroll
    for (int t = 0; t < 2; ++t) {
#pragma unroll
        for (int n = 0; n < 4; ++n) {
#pragma unroll
            for (int r = 0; r < 8; ++r) {
                const int irow = t * 16 + r + hlf * 8;
                const int i = i0 + irow;
                const float o = acc[t][n][r] / rowsum[irow];
                out[(((size_t)(b * L_ + i) * V_) + k) * D_ + n * 16 + l16] = o;
            }
        }
    }
}

// ---------------------------------------------------------------------------
extern "C" void kernel_launch(void* const* d_in, const int* in_sizes, int n_in,
                              void* d_out, int out_size, void* d_ws, size_t ws_size,
                              hipStream_t stream) {
    const float* query = (const float*)d_in[0];
    const float* key   = (const float*)d_in[1];
    const float* value = (const float*)d_in[2];
    const int*   label = (const int*)d_in[3];
    float* out  = (float*)d_out;
    float* sumv = (float*)d_ws;   // B*L*D fp32 = 512 KB scratch

    // 1) key summed over the head axis
    const int n1 = B_ * L_ * D_;
    sum_key_kernel<<<(n1 + 255) / 256, 256, 0, stream>>>(key, sumv);

    // 2) clustered attention: one wave per (b, head, 32-row tile)
    const int blocks = B_ * V_ * (L_ / MT);   // 2048
    attn_kernel<<<blocks, 32, 0, stream>>>(query, sumv, value, label, out);
}